// GCN_17600775979603
// MI455X (gfx1250) — compile-verified
//
#include <hip/hip_runtime.h>
#include <hip/hip_bf16.h>
#include <stdint.h>

// CDNA5 / gfx1250 GCN forward.
// Exact-fp32 GEMM via V_WMMA_F32_16X16X4_F32; edge aggregation via f32 atomics
// (hits L2 since t[] = 51MB < 192MB L2).

typedef __attribute__((ext_vector_type(2))) float v2f;
typedef __attribute__((ext_vector_type(8))) float v8f;

#define DIM 128
#define LDS_PITCH 130   // pad to avoid LDS bank conflicts on transposed W reads

// ---------------- norm precompute ----------------

__global__ __launch_bounds__(256) void gcn_deg_init(float* __restrict__ deg, int N) {
    int i = blockIdx.x * 256 + threadIdx.x;
    if (i < N) deg[i] = 1.0f;                       // self-loop weight
}

__global__ __launch_bounds__(256) void gcn_deg_accum(const long long* __restrict__ dst,
                                                     const float* __restrict__ ew,
                                                     float* __restrict__ deg, int E) {
    int i = blockIdx.x * 256 + threadIdx.x;
    if (i < E) atomicAdd(&deg[dst[i]], ew[i]);
}

__global__ __launch_bounds__(256) void gcn_rsqrt_inplace(float* __restrict__ d, int N) {
    int i = blockIdx.x * 256 + threadIdx.x;
    if (i < N) d[i] = rsqrtf(d[i]);                 // deg -> dinv
}

__global__ __launch_bounds__(256) void gcn_edge_norm(const long long* __restrict__ src,
                                                     const long long* __restrict__ dst,
                                                     const float* __restrict__ ew,
                                                     const float* __restrict__ dinv,
                                                     float* __restrict__ norm, int E) {
    int i = blockIdx.x * 256 + threadIdx.x;
    if (i < E) norm[i] = dinv[src[i]] * ew[i] * dinv[dst[i]];
}

__global__ __launch_bounds__(256) void gcn_copy4(const float4* __restrict__ x,
                                                 float4* __restrict__ h, int n4) {
    int i = blockIdx.x * 256 + threadIdx.x;
    if (i < n4) h[i] = x[i];
}

// ---------------- dense: t = h @ W ; agg = t*dinv^2 + bias ----------------
// 8 waves / block; wave = one 16-row tile across all 8 column tiles.
__global__ __launch_bounds__(256) void gcn_gemm_wmma(const float* __restrict__ h,
                                                     const float* __restrict__ W,
                                                     const float* __restrict__ bias,
                                                     const float* __restrict__ dinv,
                                                     float* __restrict__ t,
                                                     float* __restrict__ agg, int N) {
    __shared__ float Wl[DIM * LDS_PITCH];           // W transposed: Wl[n*PITCH + k]
    const int tid = threadIdx.x;

    // Cooperative load + transpose of the 64KB weight matrix into LDS.
    const float4* Wv = (const float4*)W;
    #pragma unroll
    for (int i = 0; i < 16; ++i) {
        int e4 = tid + i * 256;                     // float4 index, 0..4095
        int k  = e4 >> 5;                           // 32 float4 per k-row
        int n0 = (e4 & 31) << 2;
        float4 v = Wv[e4];
        Wl[(n0 + 0) * LDS_PITCH + k] = v.x;
        Wl[(n0 + 1) * LDS_PITCH + k] = v.y;
        Wl[(n0 + 2) * LDS_PITCH + k] = v.z;
        Wl[(n0 + 3) * LDS_PITCH + k] = v.w;
    }
    __syncthreads();

    const int wave    = tid >> 5;
    const int lane    = tid & 31;
    const int rowBase = (blockIdx.x * 8 + wave) * 16;
    if (rowBase >= N) return;                       // N % 16 == 0: no partial tiles
    const int half = lane >> 4;                     // lane half selects K pair
    const int l15  = lane & 15;

    v8f acc[8] = {};                                // 8 column tiles of 16x16 f32

    // A fragment: lane (half,l15) holds row rowBase+l15, K = k0 + 2*half + {0,1}
    const float* arow = h + (size_t)(rowBase + l15) * DIM;
    for (int k0 = 0; k0 < DIM; k0 += 4) {
        const int ka = k0 + half * 2;
        v2f afrag = *(const v2f*)(arow + ka);       // global_load_b64, each elt read once
        #pragma unroll
        for (int nt = 0; nt < 8; ++nt) {
            // B fragment: K = ka + {0,1}, Ncol = nt*16 + l15 (from transposed LDS)
            v2f bfrag = *(const v2f*)(&Wl[(nt * 16 + l15) * LDS_PITCH + ka]);
            acc[nt] = __builtin_amdgcn_wmma_f32_16x16x4_f32(
                false, afrag, false, bfrag, (short)0, acc[nt], false, false);
        }
    }

    // self-loop coefficients dinv^2 for this tile's rows
    float sn[8];
    #pragma unroll
    for (int r = 0; r < 8; ++r) {
        float dv = dinv[rowBase + r + half * 8];
        sn[r] = dv * dv;
    }

    // C/D layout: VGPR r -> M = r (lanes 0-15) / M = r+8 (lanes 16-31), N = l15
    #pragma unroll
    for (int nt = 0; nt < 8; ++nt) {
        const int col = nt * 16 + l15;
        const float b = bias[col];
        #pragma unroll
        for (int r = 0; r < 8; ++r) {
            const size_t idx = (size_t)(rowBase + r + half * 8) * DIM + col;
            const float v = acc[nt][r];
            t[idx]   = v;
            agg[idx] = v * sn[r] + b;               // fused self-loop + bias
        }
    }
}

// ---------------- sparse: agg[dst] += t[src] * norm ----------------
// One wave per edge; each lane handles 4 features (float4 gather + 4 f32 atomics).
__global__ __launch_bounds__(256) void gcn_scatter(const long long* __restrict__ src,
                                                   const long long* __restrict__ dst,
                                                   const float* __restrict__ norm,
                                                   const float* __restrict__ t,
                                                   float* __restrict__ agg, int E) {
    const int gwave = (blockIdx.x * 256 + threadIdx.x) >> 5;
    const int lane  = threadIdx.x & 31;
    if (gwave >= E) return;
    const long long s = src[gwave];
    const long long d = dst[gwave];
    const float nm = norm[gwave];
    const float4 v = ((const float4*)(t + (size_t)s * DIM))[lane];  // L2-resident gather
    float* ap = agg + (size_t)d * DIM + lane * 4;
    atomicAdd(ap + 0, v.x * nm);
    atomicAdd(ap + 1, v.y * nm);
    atomicAdd(ap + 2, v.z * nm);
    atomicAdd(ap + 3, v.w * nm);
}

// ---------------- h = relu(agg) + h ----------------
__global__ __launch_bounds__(256) void gcn_relu_res(const float4* __restrict__ agg,
                                                    float4* __restrict__ h, int n4) {
    int i = blockIdx.x * 256 + threadIdx.x;
    if (i < n4) {
        float4 a = agg[i];
        float4 hv = h[i];
        hv.x += fmaxf(a.x, 0.f);
        hv.y += fmaxf(a.y, 0.f);
        hv.z += fmaxf(a.z, 0.f);
        hv.w += fmaxf(a.w, 0.f);
        h[i] = hv;
    }
}

extern "C" void kernel_launch(void* const* d_in, const int* in_sizes, int n_in,
                              void* d_out, int out_size, void* d_ws, size_t ws_size,
                              hipStream_t stream) {
    const float*     x   = (const float*)d_in[0];          // [N,128] f32
    const long long* ei  = (const long long*)d_in[1];      // [2,E] i64
    const float*     ew  = (const float*)d_in[2];          // [E] f32
    const float*     Ws  = (const float*)d_in[3];          // [L,128,128] f32
    const float*     bs  = (const float*)d_in[4];          // [L,128] f32

    const int N = in_sizes[0] / DIM;
    const int E = in_sizes[2];
    const int L = in_sizes[3] / (DIM * DIM);
    const long long* esrc = ei;
    const long long* edst = ei + E;

    // workspace layout: dinv[N] | norm[E] | t[N*128] | agg[N*128]  (~110 MB)
    float* dinv = (float*)d_ws;
    float* norm = dinv + N;
    float* t    = norm + E;
    float* agg  = t + (size_t)N * DIM;
    float* h    = (float*)d_out;                           // residual stream lives in d_out

    const int n4 = N * DIM / 4;

    gcn_deg_init   <<<(N + 255) / 256, 256, 0, stream>>>(dinv, N);
    gcn_deg_accum  <<<(E + 255) / 256, 256, 0, stream>>>(edst, ew, dinv, E);
    gcn_rsqrt_inplace<<<(N + 255) / 256, 256, 0, stream>>>(dinv, N);
    gcn_edge_norm  <<<(E + 255) / 256, 256, 0, stream>>>(esrc, edst, ew, dinv, norm, E);
    gcn_copy4      <<<(n4 + 255) / 256, 256, 0, stream>>>((const float4*)x, (float4*)h, n4);

    const int rowTiles  = (N + 15) / 16;
    const int gemmBlks  = (rowTiles + 7) / 8;
    const int scatBlks  = (E + 7) / 8;                     // 8 waves (= 8 edges) per block

    for (int l = 0; l < L; ++l) {
        gcn_gemm_wmma<<<gemmBlks, 256, 0, stream>>>(h, Ws + (size_t)l * DIM * DIM,
                                                    bs + (size_t)l * DIM, dinv, t, agg, N);
        gcn_scatter  <<<scatBlks, 256, 0, stream>>>(esrc, edst, norm, t, agg, E);
        gcn_relu_res <<<(n4 + 255) / 256, 256, 0, stream>>>((const float4*)agg, (float4*)h, n4);
    }
}